// GAT_40218073760152
// MI455X (gfx1250) — compile-verified
//
#include <hip/hip_runtime.h>
#include <hip/hip_bf16.h>

// ---------------------------------------------------------------------------
// 2-layer GAT for MI455X (gfx1250, wave32).
// Dense GEMMs use V_WMMA_F32_16X16X4_F32 (exact f32 matrix pipe).
// Edge softmax/aggregation uses f32 atomics (monotone-uint max trick).
// ---------------------------------------------------------------------------

typedef float v2f __attribute__((ext_vector_type(2)));
typedef float v8f __attribute__((ext_vector_type(8)));

#define NEG_SLOPE 0.2f

// ------------------------- WMMA f32 GEMM -----------------------------------
// C[M,NCOLS] = A[M,K] * B[K,NCOLS], row-major, K/NCOLS compile-time.
// Requires M % 16 == 0, K % 4 == 0. Each wave computes a 16 x (16*NTILES)
// strip: one A fragment feeds NTILES WMMAs (independent accumulator chains,
// 4x less A traffic, fully-immediate B offsets). Narrow C (NCOLS < 16,
// MASKED=true) handled by clamped loads + zero multiply -- no EXEC
// divergence around the WMMAs. blockDim.x = 128 (4 waves).
template <int NTILES, int K, int NCOLS, bool MASKED>
__global__ void gat_wmma_gemm_f32(const float* __restrict__ A,
                                  const float* __restrict__ B,
                                  float* __restrict__ C,
                                  int M) {
    const int wave = threadIdx.x >> 5;        // wave32
    const int lane = threadIdx.x & 31;
    const int MT = M >> 4;
    const int mt = blockIdx.x * 4 + wave;     // uniform across the wave
    if (mt >= MT) return;                     // whole-wave exit -> EXEC all-1s below

    const int m0 = mt << 4;
    const int n0 = blockIdx.y * (16 * NTILES);

    const int lrow  = lane & 15;
    const int khalf = (lane >> 4) << 1;       // 0 or 2 (ISA 32-bit A/B striping)

    // A: per-lane float2 at (m0+lrow, k+khalf), stepped by 4 floats per K-step
    const float* ap = A + (size_t)(m0 + lrow) * (size_t)K + khalf;

    // B: one base pointer; tile j's pair at immediate offsets {16j, 16j+NCOLS}
    const int col0 = n0 + lrow;
    int colc = col0;
    float bmask = 1.0f;
    if (MASKED) {                             // narrow-C: clamp + zero mask
        colc  = (col0 < NCOLS) ? col0 : (NCOLS - 1);
        bmask = (col0 < NCOLS) ? 1.0f : 0.0f;
    }
    const float* bp = B + khalf * NCOLS + colc;

    v8f acc[NTILES];
    #pragma unroll
    for (int j = 0; j < NTILES; ++j)
        acc[j] = (v8f){0.f, 0.f, 0.f, 0.f, 0.f, 0.f, 0.f, 0.f};

    #pragma unroll 8
    for (int k = 0; k < K; k += 4) {
        v2f a = *(const v2f*)ap;              // 8B-aligned (K even, khalf even)
        ap += 4;
        #pragma unroll
        for (int j = 0; j < NTILES; ++j) {
            v2f b;
            b.x = bp[16 * j];
            b.y = bp[16 * j + NCOLS];
            if (MASKED) { b.x *= bmask; b.y *= bmask; }
            acc[j] = __builtin_amdgcn_wmma_f32_16x16x4_f32(
                         /*neg_a=*/false, a, /*neg_b=*/false, b,
                         /*c_mod=*/(short)0, acc[j],
                         /*reuse_a=*/false, /*reuse_b=*/false);
        }
        bp += 4 * NCOLS;
    }

    // C/D layout: VGPR r -> row m0 + (lane>=16 ? 8 : 0) + r, col n0+16j+(lane&15)
    const int srow = m0 + ((lane >> 4) << 3);
    #pragma unroll
    for (int j = 0; j < NTILES; ++j) {
        const int scol = n0 + 16 * j + lrow;
        if (!MASKED || scol < NCOLS) {
            #pragma unroll
            for (int r = 0; r < 8; ++r)
                C[(size_t)(srow + r) * (size_t)NCOLS + scol] = acc[j][r];
        }
    }
}

// ------------------------- helpers -----------------------------------------
__device__ __forceinline__ void edge_sd(const long long* __restrict__ ei,
                                        int E, int i, int& s, int& d) {
    if (i < E) { s = (int)ei[i]; d = (int)ei[E + i]; }
    else       { s = i - E; d = i - E; }          // implicit self-loop
}

__device__ __forceinline__ float lrelu(float v) {
    return v > 0.f ? v : NEG_SLOPE * v;
}

// order-preserving f32 <-> u32 encoding for atomic max
__device__ __forceinline__ unsigned enc_f32(float f) {
    unsigned u = __float_as_uint(f);
    return (u & 0x80000000u) ? ~u : (u | 0x80000000u);
}
__device__ __forceinline__ float dec_f32(unsigned u) {
    return (u & 0x80000000u) ? __uint_as_float(u & 0x7FFFFFFFu)
                             : __uint_as_float(~u);
}

// ------------------------- per-node attention logits ------------------------
__global__ void gat_node_logits(const float* __restrict__ h,
                                const float* __restrict__ a_src,
                                const float* __restrict__ a_dst,
                                float* __restrict__ als,
                                float* __restrict__ ald,
                                int Nn, int H, int F) {
    int idx = blockIdx.x * blockDim.x + threadIdx.x;
    if (idx >= Nn * H) return;
    int n = idx / H, hh = idx - n * H;
    const float* hp = h + (size_t)n * (size_t)(H * F) + (size_t)hh * F;
    const float* as = a_src + hh * F;
    const float* ad = a_dst + hh * F;
    float s = 0.f, d = 0.f;
    for (int f = 0; f < F; ++f) { s += hp[f] * as[f]; d += hp[f] * ad[f]; }
    als[idx] = s;
    ald[idx] = d;
}

// ------------------------- edge passes -------------------------------------
__global__ void gat_edge_max(const long long* __restrict__ ei, int E, int ET,
                             const float* __restrict__ als,
                             const float* __restrict__ ald,
                             unsigned* __restrict__ menc, int H) {
    int idx = blockIdx.x * blockDim.x + threadIdx.x;
    if (idx >= ET * H) return;
    int e = idx / H, hh = idx - e * H;
    int s, d; edge_sd(ei, E, e, s, d);
    float v = lrelu(als[s * H + hh] + ald[d * H + hh]);
    atomicMax(&menc[d * H + hh], enc_f32(v));
}

__global__ void gat_decode_max(unsigned* __restrict__ menc, int count) {
    int idx = blockIdx.x * blockDim.x + threadIdx.x;
    if (idx >= count) return;
    float m = dec_f32(menc[idx]);            // every node has a self-loop -> finite
    ((float*)menc)[idx] = m;
}

__global__ void gat_edge_expsum(const long long* __restrict__ ei, int E, int ET,
                                const float* __restrict__ als,
                                const float* __restrict__ ald,
                                const float* __restrict__ m,
                                float* __restrict__ den, int H) {
    int idx = blockIdx.x * blockDim.x + threadIdx.x;
    if (idx >= ET * H) return;
    int e = idx / H, hh = idx - e * H;
    int s, d; edge_sd(ei, E, e, s, d);
    float v = lrelu(als[s * H + hh] + ald[d * H + hh]);
    atomicAdd(&den[d * H + hh], expf(v - m[d * H + hh]));
}

// alpha[e, h] = exp(lrelu(e_logit) - max[dst]) / (den[dst] + eps)
// Materialized once so the (edge x feature) scatter below is pure load+atomic.
__global__ void gat_edge_alpha(const long long* __restrict__ ei, int E, int ET,
                               const float* __restrict__ als,
                               const float* __restrict__ ald,
                               const float* __restrict__ m,
                               const float* __restrict__ den,
                               float* __restrict__ alpha, int H) {
    int idx = blockIdx.x * blockDim.x + threadIdx.x;
    if (idx >= ET * H) return;
    int e = idx / H, hh = idx - e * H;
    int s, d; edge_sd(ei, E, e, s, d);
    int id = d * H + hh;
    float v = lrelu(als[s * H + hh] + ald[id]);
    alpha[idx] = expf(v - m[id]) / (den[id] + 1e-16f);
}

__global__ void gat_edge_agg(const long long* __restrict__ ei, int E, int ET,
                             const float* __restrict__ alpha,
                             const float* __restrict__ h,
                             float* __restrict__ agg, int H, int F) {
    const int HF = H * F;
    long long gid = (long long)blockIdx.x * blockDim.x + threadIdx.x;
    if (gid >= (long long)ET * HF) return;
    int e = (int)(gid / HF);
    int c = (int)(gid - (long long)e * HF);
    int hh = c / F;
    int s, d; edge_sd(ei, E, e, s, d);
    atomicAdd(&agg[(size_t)d * HF + c],
              alpha[(size_t)e * H + hh] * h[(size_t)s * HF + c]);
}

// ------------------------- epilogues ---------------------------------------
__global__ void gat_bias_elu(float* __restrict__ io, const float* __restrict__ b,
                             int Nn, int C) {
    int idx = blockIdx.x * blockDim.x + threadIdx.x;
    if (idx >= Nn * C) return;
    int c = idx % C;
    float v = io[idx] + b[c];
    io[idx] = v > 0.f ? v : (expf(v) - 1.0f);   // ELU(alpha=1)
}

__global__ void gat_final_linear_sigmoid(const float* __restrict__ h,
                                         const float* __restrict__ Wl,
                                         const float* __restrict__ bl,
                                         float* __restrict__ out, int Nn) {
    int n = blockIdx.x * blockDim.x + threadIdx.x;
    if (n >= Nn) return;
    const float* hp = h + (size_t)n * 8;
    #pragma unroll
    for (int o = 0; o < 2; ++o) {
        float acc = bl[o];
        #pragma unroll
        for (int f = 0; f < 8; ++f) acc += hp[f] * Wl[f * 2 + o];
        out[(size_t)n * 2 + o] = 1.0f / (1.0f + expf(-acc));
    }
}

// ------------------------- launch ------------------------------------------
static inline unsigned cdiv(long long a, long long b) { return (unsigned)((a + b - 1) / b); }

extern "C" void kernel_launch(void* const* d_in, const int* in_sizes, int n_in,
                              void* d_out, int out_size, void* d_ws, size_t ws_size,
                              hipStream_t stream) {
    const float*      x   = (const float*)d_in[0];
    const long long*  ei  = (const long long*)d_in[1];   // [2, E] int64
    const float*      W1  = (const float*)d_in[3];       // [128, 128]
    const float*      as1 = (const float*)d_in[4];       // [4, 32]
    const float*      ad1 = (const float*)d_in[5];
    const float*      b1  = (const float*)d_in[6];       // [128]
    const float*      W2  = (const float*)d_in[7];       // [128, 8]
    const float*      as2 = (const float*)d_in[8];       // [1, 8]
    const float*      ad2 = (const float*)d_in[9];
    const float*      b2  = (const float*)d_in[10];      // [8]
    const float*      Wl  = (const float*)d_in[11];      // [8, 2]
    const float*      bl  = (const float*)d_in[12];      // [2]
    float*            out = (float*)d_out;

    const int IN = 128, H1 = 4, F1 = 32, C1 = 128, F2 = 8;
    const int Nn = in_sizes[0] / IN;          // 50000
    const int E  = in_sizes[1] / 2;           // 800000
    const int ET = E + Nn;                    // + self loops

    // workspace carve-up (256B aligned)
    char* ws = (char*)d_ws;
    size_t off = 0;
    auto carve = [&](size_t bytes) -> char* {
        char* p = ws + off;
        off += (bytes + 255) & ~(size_t)255;
        return p;
    };
    float*    h1raw  = (float*)carve((size_t)Nn * C1 * 4);
    float*    agg1   = (float*)carve((size_t)Nn * C1 * 4);   // becomes h1 (bias+ELU)
    float*    als1   = (float*)carve((size_t)Nn * H1 * 4);
    float*    ald1   = (float*)carve((size_t)Nn * H1 * 4);
    unsigned* m1     = (unsigned*)carve((size_t)Nn * H1 * 4);
    float*    den1   = (float*)carve((size_t)Nn * H1 * 4);
    float*    alpha1 = (float*)carve((size_t)ET * H1 * 4);
    float*    h2raw  = (float*)carve((size_t)Nn * F2 * 4);
    float*    agg2   = (float*)carve((size_t)Nn * F2 * 4);   // becomes h2
    float*    als2   = (float*)carve((size_t)Nn * 4);
    float*    ald2   = (float*)carve((size_t)Nn * 4);
    unsigned* m2     = (unsigned*)carve((size_t)Nn * 4);
    float*    den2   = (float*)carve((size_t)Nn * 4);
    float*    alpha2 = (float*)carve((size_t)ET * 4);

    // zero accumulators (enc(-inf) lower bound == 0 for the max buffers)
    hipMemsetAsync(agg1, 0, (size_t)Nn * C1 * 4, stream);
    hipMemsetAsync(m1,   0, (size_t)Nn * H1 * 4, stream);
    hipMemsetAsync(den1, 0, (size_t)Nn * H1 * 4, stream);
    hipMemsetAsync(agg2, 0, (size_t)Nn * F2 * 4, stream);
    hipMemsetAsync(m2,   0, (size_t)Nn * 4, stream);
    hipMemsetAsync(den2, 0, (size_t)Nn * 4, stream);

    const int TB = 256;
    const unsigned gemmBlocks = cdiv(Nn / 16, 4);   // 4 waves (M tiles) per block

    // ---------------- layer 1 ----------------
    // h1raw = x @ W1 : each wave does 16x64 strip (4 WMMA tiles), 2 col groups
    gat_wmma_gemm_f32<4, 128, 128, false>
        <<<dim3(gemmBlocks, C1 / 64), 128, 0, stream>>>(x, W1, h1raw, Nn);
    gat_node_logits<<<cdiv((long long)Nn * H1, TB), TB, 0, stream>>>(h1raw, as1, ad1, als1, ald1, Nn, H1, F1);
    gat_edge_max<<<cdiv((long long)ET * H1, TB), TB, 0, stream>>>(ei, E, ET, als1, ald1, m1, H1);
    gat_decode_max<<<cdiv((long long)Nn * H1, TB), TB, 0, stream>>>(m1, Nn * H1);
    gat_edge_expsum<<<cdiv((long long)ET * H1, TB), TB, 0, stream>>>(ei, E, ET, als1, ald1, (float*)m1, den1, H1);
    gat_edge_alpha<<<cdiv((long long)ET * H1, TB), TB, 0, stream>>>(ei, E, ET, als1, ald1, (float*)m1, den1, alpha1, H1);
    gat_edge_agg<<<cdiv((long long)ET * C1, TB), TB, 0, stream>>>(ei, E, ET, alpha1, h1raw, agg1, H1, F1);
    gat_bias_elu<<<cdiv((long long)Nn * C1, TB), TB, 0, stream>>>(agg1, b1, Nn, C1);  // agg1 = h1

    // ---------------- layer 2 (H=1, F=8; mean over 1 head == identity) ------
    // h2raw = h1 @ W2 : single masked column tile (Ncols=8 < 16)
    gat_wmma_gemm_f32<1, 128, 8, true>
        <<<dim3(gemmBlocks, 1), 128, 0, stream>>>(agg1, W2, h2raw, Nn);
    gat_node_logits<<<cdiv((long long)Nn, TB), TB, 0, stream>>>(h2raw, as2, ad2, als2, ald2, Nn, 1, F2);
    gat_edge_max<<<cdiv((long long)ET, TB), TB, 0, stream>>>(ei, E, ET, als2, ald2, m2, 1);
    gat_decode_max<<<cdiv((long long)Nn, TB), TB, 0, stream>>>(m2, Nn);
    gat_edge_expsum<<<cdiv((long long)ET, TB), TB, 0, stream>>>(ei, E, ET, als2, ald2, (float*)m2, den2, 1);
    gat_edge_alpha<<<cdiv((long long)ET, TB), TB, 0, stream>>>(ei, E, ET, als2, ald2, (float*)m2, den2, alpha2, 1);
    gat_edge_agg<<<cdiv((long long)ET * F2, TB), TB, 0, stream>>>(ei, E, ET, alpha2, h2raw, agg2, 1, F2);
    gat_bias_elu<<<cdiv((long long)Nn * F2, TB), TB, 0, stream>>>(agg2, b2, Nn, F2);  // agg2 = h2

    // ---------------- final linear + sigmoid --------------------------------
    gat_final_linear_sigmoid<<<cdiv((long long)Nn, TB), TB, 0, stream>>>(agg2, Wl, bl, out, Nn);
}